// SiamNet_75625784148568
// MI455X (gfx1250) — compile-verified
//
#include <hip/hip_runtime.h>
#include <hip/hip_bf16.h>
#include <math.h>

#define NNODES 100000
#define NFEAT  128
#define HID    16
#define NCLS   10
#define NTILES (NNODES / 16)   // 6250, exact

typedef __attribute__((ext_vector_type(2))) float v2f;
typedef __attribute__((ext_vector_type(8))) float v8f;

// ---------------------------------------------------------------------------
// Kernel 1: xw[n, 0:32] = x[n,:] @ [W1 | W2]   using V_WMMA_F32_16X16X4_F32
// One wave (32 lanes) computes one 16x32 output tile; 4 waves / block.
// A (16x128) staged in LDS via coalesced float4 loads.
// W stored TRANSPOSED in LDS ([n][k], padded row = 132 floats) so each lane's
// B K-pair is a contiguous 8-byte LDS read (ds_load_b64), like the A operand.
// ---------------------------------------------------------------------------
__global__ __launch_bounds__(128) void k_gemm_xw(const float* __restrict__ x,
                                                 const float* __restrict__ W1,
                                                 const float* __restrict__ W2,
                                                 float* __restrict__ xw) {
    __shared__ float sWt[32][132];      // [n][k] transposed weights, ~16.9 KB
    __shared__ float sA[4][16][132];    // per-wave 16x128 A tile (+pad), ~33.8 KB

    const int lane = threadIdx.x & 31;
    const int wave = threadIdx.x >> 5;

    // Load [W1|W2] transposed: sWt[n][k] = Wcat[k][n]
    for (int i = threadIdx.x; i < NFEAT * 32; i += 128) {
        int k = i >> 5, n = i & 31;
        float w = (n < HID) ? W1[k * HID + n] : W2[k * HID + (n - HID)];
        sWt[n][k] = w;
    }

    const int tile = blockIdx.x * 4 + wave;
    const bool active = tile < NTILES;
    if (active) {
        const int rbase = tile * 16;
        const float4* xr = (const float4*)x;   // x row = 32 float4
        for (int i = lane; i < 16 * 32; i += 32) {
            int row = i >> 5, c4 = i & 31;
            float4 v = xr[(size_t)(rbase + row) * 32 + c4];
            sA[wave][row][c4 * 4 + 0] = v.x;
            sA[wave][row][c4 * 4 + 1] = v.y;
            sA[wave][row][c4 * 4 + 2] = v.z;
            sA[wave][row][c4 * 4 + 3] = v.w;
        }
    }
    __syncthreads();
    if (!active) return;

    v8f acc0 = {};   // N-tile 0  (W1 half)
    v8f acc1 = {};   // N-tile 1  (W2 half)
    const int m    = lane & 15;     // M (for A) and N (for B/C)
    const int half = lane >> 4;     // K-pair select / row-half select

    const float* arow  = &sA[wave][m][0];
    const float* brow0 = &sWt[m][0];
    const float* brow1 = &sWt[m + 16][0];

    for (int k = 0; k < NFEAT; k += 4) {
        int k2 = k + 2 * half;                       // even -> 8B aligned
        v2f a  = *(const v2f*)(arow  + k2);
        v2f b0 = *(const v2f*)(brow0 + k2);
        v2f b1 = *(const v2f*)(brow1 + k2);
        acc0 = __builtin_amdgcn_wmma_f32_16x16x4_f32(false, a, false, b0,
                                                     (short)0, acc0, false, false);
        acc1 = __builtin_amdgcn_wmma_f32_16x16x4_f32(false, a, false, b1,
                                                     (short)0, acc1, false, false);
    }

    const int rbase = tile * 16;
    for (int j = 0; j < 8; ++j) {
        int row = rbase + j + half * 8;            // C/D layout: VGPR j -> M = j + 8*half
        xw[(size_t)row * 32 + m]      = acc0[j];
        xw[(size_t)row * 32 + 16 + m] = acc1[j];
    }
}

// ---------------------------------------------------------------------------
// Degree / norm kernels. deg includes self-loop (init 1.0).
// ---------------------------------------------------------------------------
__global__ void k_init_deg(float* d1, float* d2, int n) {
    int i = blockIdx.x * blockDim.x + threadIdx.x;
    if (i < n) { d1[i] = 1.0f; d2[i] = 1.0f; }
}

__global__ void k_count_deg(const int* __restrict__ dst, int E, float* __restrict__ deg) {
    int e = blockIdx.x * blockDim.x + threadIdx.x;
    if (e < E) atomicAdd(&deg[dst[e]], 1.0f);
}

__global__ void k_rsqrt(float* d, int n) {
    int i = blockIdx.x * blockDim.x + threadIdx.x;
    if (i < n) d[i] = rsqrtf(d[i]);   // deg >= 1 always (self-loop)
}

// ---------------------------------------------------------------------------
// agg[n,c] = xw[n,c] * dinv^2   (self-loop contribution; also zero-inits agg)
// ---------------------------------------------------------------------------
__global__ void k_self_init(const float* __restrict__ xw,
                            const float* __restrict__ dinv1,
                            const float* __restrict__ dinv2,
                            float* __restrict__ agg) {
    int i = blockIdx.x * blockDim.x + threadIdx.x;
    if (i >= NNODES * 32) return;
    int n = i >> 5, c = i & 31;
    float di = (c < HID) ? dinv1[n] : dinv2[n];
    agg[i] = xw[i] * di * di;
}

// ---------------------------------------------------------------------------
// Edge scatter for one branch: 4 threads/edge, each handles 4 features.
// ---------------------------------------------------------------------------
__global__ void k_scatter16(const int* __restrict__ src, const int* __restrict__ dst,
                            int E, const float* __restrict__ dinv,
                            const float* __restrict__ xw, float* __restrict__ agg,
                            int colbase) {
    int idx = blockIdx.x * blockDim.x + threadIdx.x;
    if (idx >= E * 4) return;
    int e = idx >> 2, g = idx & 3;
    int s = src[e], d = dst[e];
    float w = dinv[s] * dinv[d];
    float4 v = ((const float4*)xw)[(size_t)s * 8 + (colbase >> 2) + g];
    float* out = agg + (size_t)d * 32 + colbase + g * 4;
    atomicAdd(out + 0, v.x * w);
    atomicAdd(out + 1, v.y * w);
    atomicAdd(out + 2, v.z * w);
    atomicAdd(out + 3, v.w * w);
}

__global__ void k_relu_bias(float* __restrict__ h, const float* __restrict__ b1,
                            const float* __restrict__ b2) {
    int i = blockIdx.x * blockDim.x + threadIdx.x;
    if (i >= NNODES * 32) return;
    int c = i & 31;
    float bb = (c < HID) ? b1[c] : b2[c - HID];
    h[i] = fmaxf(h[i] + bb, 0.0f);
}

// ---------------------------------------------------------------------------
// hw[n,0:10] = h[n,0:32] @ W4   (tiny GEMM, W4 in LDS, float4 row loads)
// ---------------------------------------------------------------------------
__global__ __launch_bounds__(256) void k_gemm2(const float* __restrict__ h,
                                               const float* __restrict__ W4,
                                               float* __restrict__ hw) {
    __shared__ float sW[32 * NCLS];
    for (int i = threadIdx.x; i < 32 * NCLS; i += 256) sW[i] = W4[i];
    __syncthreads();
    int n = blockIdx.x * 256 + threadIdx.x;
    if (n >= NNODES) return;
    float acc[NCLS];
    for (int j = 0; j < NCLS; ++j) acc[j] = 0.0f;
    const float4* hr = (const float4*)(h + (size_t)n * 32);
    for (int q = 0; q < 8; ++q) {
        float4 v = hr[q];
        int k = q * 4;
        for (int j = 0; j < NCLS; ++j)
            acc[j] += v.x * sW[(k + 0) * NCLS + j] + v.y * sW[(k + 1) * NCLS + j] +
                      v.z * sW[(k + 2) * NCLS + j] + v.w * sW[(k + 3) * NCLS + j];
    }
    for (int j = 0; j < NCLS; ++j) hw[(size_t)n * NCLS + j] = acc[j];
}

__global__ void k_self_init3(const float* __restrict__ hw, const float* __restrict__ dinv1,
                             float* __restrict__ out) {
    int i = blockIdx.x * blockDim.x + threadIdx.x;
    if (i >= NNODES * NCLS) return;
    int n = i / NCLS;
    float w = dinv1[n] * dinv1[n];
    out[i] = hw[i] * w;
}

__global__ void k_scatter3(const int* __restrict__ src, const int* __restrict__ dst,
                           int E, const float* __restrict__ dinv,
                           const float* __restrict__ hw, float* __restrict__ out) {
    int e = blockIdx.x * blockDim.x + threadIdx.x;
    if (e >= E) return;
    int s = src[e], d = dst[e];
    float w = dinv[s] * dinv[d];
    const float* hs = hw + (size_t)s * NCLS;
    float* od = out + (size_t)d * NCLS;
    for (int j = 0; j < NCLS; ++j) atomicAdd(od + j, hs[j] * w);
}

__global__ void k_logsoftmax(float* __restrict__ out, const float* __restrict__ b4) {
    int n = blockIdx.x * blockDim.x + threadIdx.x;
    if (n >= NNODES) return;
    float v[NCLS];
    float m = -INFINITY;
    for (int j = 0; j < NCLS; ++j) {
        v[j] = out[(size_t)n * NCLS + j] + b4[j];
        m = fmaxf(m, v[j]);
    }
    float s = 0.0f;
    for (int j = 0; j < NCLS; ++j) s += expf(v[j] - m);
    float lse = m + logf(s);
    for (int j = 0; j < NCLS; ++j) out[(size_t)n * NCLS + j] = v[j] - lse;
}

// ---------------------------------------------------------------------------
extern "C" void kernel_launch(void* const* d_in, const int* in_sizes, int n_in,
                              void* d_out, int out_size, void* d_ws, size_t ws_size,
                              hipStream_t stream) {
    const float* x  = (const float*)d_in[0];
    const int*   e1 = (const int*)d_in[1];       // [2, E1]: src then dst
    const int*   e2 = (const int*)d_in[2];
    const float* W1 = (const float*)d_in[3];
    const float* b1 = (const float*)d_in[4];
    const float* W2 = (const float*)d_in[5];
    const float* b2 = (const float*)d_in[6];
    const float* W4 = (const float*)d_in[7];
    const float* b4 = (const float*)d_in[8];
    float* out = (float*)d_out;

    const int E1 = in_sizes[1] / 2, E2 = in_sizes[2] / 2;
    const int* src1 = e1;  const int* dst1 = e1 + E1;
    const int* src2 = e2;  const int* dst2 = e2 + E2;

    // workspace layout (floats): xw[32N] | dinv1[N] | dinv2[N] | agg[32N] | hw[10N]
    float* xw    = (float*)d_ws;
    float* dinv1 = xw + (size_t)32 * NNODES;
    float* dinv2 = dinv1 + NNODES;
    float* agg   = dinv2 + NNODES;
    float* hw    = agg + (size_t)32 * NNODES;

    const int T = 256;
    // 1. fused projection x @ [W1|W2] via f32 WMMA
    k_gemm_xw<<<(NTILES + 3) / 4, 128, 0, stream>>>(x, W1, W2, xw);
    // 2. degrees (self-loop = 1.0) for both edge sets, then rsqrt in place
    k_init_deg<<<(NNODES + T - 1) / T, T, 0, stream>>>(dinv1, dinv2, NNODES);
    k_count_deg<<<(E1 + T - 1) / T, T, 0, stream>>>(dst1, E1, dinv1);
    k_count_deg<<<(E2 + T - 1) / T, T, 0, stream>>>(dst2, E2, dinv2);
    k_rsqrt<<<(NNODES + T - 1) / T, T, 0, stream>>>(dinv1, NNODES);
    k_rsqrt<<<(NNODES + T - 1) / T, T, 0, stream>>>(dinv2, NNODES);
    // 3. self-loop init + edge scatter (branch1 -> cols 0:16, branch2 -> 16:32)
    k_self_init<<<(NNODES * 32 + T - 1) / T, T, 0, stream>>>(xw, dinv1, dinv2, agg);
    k_scatter16<<<(E1 * 4 + T - 1) / T, T, 0, stream>>>(src1, dst1, E1, dinv1, xw, agg, 0);
    k_scatter16<<<(E2 * 4 + T - 1) / T, T, 0, stream>>>(src2, dst2, E2, dinv2, xw, agg, 16);
    // 4. bias + relu -> h (in place)
    k_relu_bias<<<(NNODES * 32 + T - 1) / T, T, 0, stream>>>(agg, b1, b2);
    // 5. layer 3: h @ W4, self-loop init into d_out, scatter, bias+log_softmax
    k_gemm2<<<(NNODES + 255) / 256, 256, 0, stream>>>(agg, W4, hw);
    k_self_init3<<<(NNODES * NCLS + T - 1) / T, T, 0, stream>>>(hw, dinv1, out);
    k_scatter3<<<(E1 + T - 1) / T, T, 0, stream>>>(src1, dst1, E1, dinv1, hw, out);
    k_logsoftmax<<<(NNODES + T - 1) / T, T, 0, stream>>>(out, b4);
}